// UKF_Aliter_1271310319799
// MI455X (gfx1250) — compile-verified
//
#include <hip/hip_runtime.h>
#include <hip/hip_bf16.h>

typedef __attribute__((ext_vector_type(2))) float v2f;
typedef __attribute__((ext_vector_type(8))) float v8f;

#define NS 32
#define TT 256
#define LDSTR 33

// Merwe weights for n=32, alpha=0.1, beta=2, kappa=-1
constexpr float DTC  = 0.02f;
constexpr float LAMW = 0.1f * 0.1f * (32.0f - 1.0f) - 32.0f;   // -31.69
constexpr float CW   = 0.5f / (32.0f + LAMW);                  // 0.5/0.31
constexpr float WM0  = LAMW / (32.0f + LAMW);
constexpr float WC0  = WM0 + (1.0f - 0.01f + 2.0f);
constexpr float SQL  = 0.55677643628300219f;                   // sqrt(0.31)

__device__ __forceinline__ float wmw(int k) { return (k == 0) ? WM0 : CW; }
__device__ __forceinline__ float wcw(int k) { return (k == 0) ? WC0 : CW; }

// Branch-free, division-free tanh: v_exp_f32 + v_rcp_f32 only.
// Clamp keeps exp finite; |x|>=15 already saturates f32 tanh to +/-1.
__device__ __forceinline__ float fast_tanh(float x) {
  float xc = fminf(15.0f, fmaxf(-15.0f, x));
  float e  = __expf(2.0f * xc);                    // v_exp_f32
  float r  = __builtin_amdgcn_rcpf(e + 1.0f);      // v_rcp_f32 (~1 ulp)
  return (e - 1.0f) * r;
}

// 1/x via v_rcp_f32 + one Newton-Raphson step (used on the serial chol chain).
__device__ __forceinline__ float fast_rcp(float x) {
  float r = __builtin_amdgcn_rcpf(x);
  return r * (2.0f - x * r);
}

// Native fp32 CDNA5 matrix op: D(16x16,f32) = A(16x4,f32) x B(4x16,f32) + C
__device__ __forceinline__ v8f wmma4(v2f a, v2f b, v8f c) {
  return __builtin_amdgcn_wmma_f32_16x16x4_f32(false, a, false, b, (short)0, c,
                                               false, false);
}

// Cooperative in-place lower Cholesky of a 32x32 (stride LDSTR) LDS matrix.
// Writes 1/diag into invd. 128 threads; trailing update is wave-strided rows x
// lane-strided columns (no integer div/mod, LDS-broadcast of the pivot column).
__device__ __forceinline__ void chol32(float* __restrict__ A,
                                       float* __restrict__ invd, int tid) {
  const int wv = tid >> 5;
  const int ln = tid & 31;
  for (int j = 0; j < NS; ++j) {
    if (tid == 0) {
      float d = __builtin_amdgcn_sqrtf(A[j * LDSTR + j]);  // v_sqrt_f32
      A[j * LDSTR + j] = d;
      invd[j] = fast_rcp(d);
    }
    __syncthreads();
    if (tid > j && tid < NS) A[tid * LDSTR + j] *= invd[j];
    __syncthreads();
    for (int i = j + 1 + wv; i < NS; i += 4) {
      float lij = A[i * LDSTR + j];
      for (int k = j + 1 + ln; k <= i; k += 32)
        A[i * LDSTR + k] -= lij * A[k * LDSTR + j];
    }
    __syncthreads();
  }
}

__global__ __launch_bounds__(128, 4) void ukf_kernel(
    const float* __restrict__ Yg, const float* __restrict__ Fg,
    const float* __restrict__ Hg, const float* __restrict__ Qg,
    const float* __restrict__ Rg, float* __restrict__ traj,
    float* __restrict__ pk, int nTraj) {
  // LDS-resident filter state (~47.5 KB / block; 320 KB per WGP available)
  __shared__ float SIGSH[65 * LDSTR];  // sigma points; reused as SH = SF @ H^T
  __shared__ float SF[65 * LDSTR];     // propagated sigma points
  __shared__ float Pm[NS * LDSTR];     // carried covariance P
  __shared__ float Pp[NS * LDSTR];
  __shared__ float Pz[NS * LDSTR];     // innovation cov; chol'd in place
  __shared__ float Pxz[NS * LDSTR];
  __shared__ float Km[NS * LDSTR];     // Kalman gain
  __shared__ float Fl[NS * LDSTR];
  __shared__ float Hl[NS * LDSTR];
  __shared__ float xv[NS], xp[NS], zp[NS], txs[NS], resid[NS], invd[NS];

  const int b = blockIdx.x;
  if (b >= nTraj) return;
  const int tid  = threadIdx.x;
  const int wave = tid >> 5;
  const int lane = tid & 31;
  const int ml   = lane & 15;        // A/B fragment M/N index
  const int half = lane >> 4;
  const int koff = half << 1;        // K sub-offset per half-wave (f32 layout)

  // ---- one-time init: stage F/H, x0=1, P0=1e-5 I, zero t=0 output slices ----
  for (int i = tid; i < NS * NS; i += 128) {
    int r = i >> 5, c = i & 31;
    Fl[r * LDSTR + c] = Fg[i];
    Hl[r * LDSTR + c] = Hg[i];
    Pm[r * LDSTR + c] = (r == c) ? 1e-5f : 0.0f;
    pk[(size_t)b * 257 * (NS * NS) + i] = 0.0f;
  }
  if (tid < NS) {
    xv[tid] = 1.0f;
    traj[(size_t)b * 257 * NS + tid] = 0.0f;
  }
  __syncthreads();

  for (int t = 0; t < TT; ++t) {
    // ---- 1) chol(P) in place (P is dead after sigma points are built) ----
    chol32(Pm, invd, tid);

    // ---- 2) sigma points rows 1..64: x +/- sqrt(n+lam)*L[:,i]; tanh(x) ----
    for (int idx = tid; idx < NS * NS; idx += 128) {
      int i = idx >> 5, j = idx & 31;
      float l = (j >= i) ? Pm[j * LDSTR + i] : 0.0f;  // lower triangle only
      float u = SQL * l;
      SIGSH[(1 + i) * LDSTR + j]  = xv[j] + u;
      SIGSH[(33 + i) * LDSTR + j] = xv[j] - u;
    }
    if (tid < NS) txs[tid] = fast_tanh(xv[tid]);
    __syncthreads();

    // ---- 3) SF = SIG + DT * tanh(SIG) @ F^T  (rows 1..64 WMMA, row 0 scalar)
    for (int tt = wave; tt < 8; tt += 4) {
      const int rowbase = 1 + 16 * (tt >> 1);
      const int colbase = 16 * (tt & 1);
      v8f acc = {};
      for (int k0 = 0; k0 < NS; k0 += 4) {
        const int k = k0 + koff;
        v2f a, bb;
        a.x  = fast_tanh(SIGSH[(rowbase + ml) * LDSTR + k]);
        a.y  = fast_tanh(SIGSH[(rowbase + ml) * LDSTR + k + 1]);
        bb.x = Fl[(colbase + ml) * LDSTR + k];      // B[k][n] = F[n][k]
        bb.y = Fl[(colbase + ml) * LDSTR + k + 1];
        acc = wmma4(a, bb, acc);
      }
      for (int v = 0; v < 8; ++v) {
        int row = rowbase + v + 8 * half;
        int col = colbase + ml;
        SF[row * LDSTR + col] = SIGSH[row * LDSTR + col] + DTC * acc[v];
      }
    }
    if (tid < NS) {  // center sigma point
      float s = 0.0f;
      for (int k = 0; k < NS; ++k) s += txs[k] * Fl[tid * LDSTR + k];
      SF[tid] = xv[tid] + DTC * s;
    }
    __syncthreads();

    // ---- 4) SH = SF @ H^T  (written over SIGSH, which is now dead) ----
    for (int tt = wave; tt < 8; tt += 4) {
      const int rowbase = 1 + 16 * (tt >> 1);
      const int colbase = 16 * (tt & 1);
      v8f acc = {};
      for (int k0 = 0; k0 < NS; k0 += 4) {
        const int k = k0 + koff;
        v2f a, bb;
        a.x  = SF[(rowbase + ml) * LDSTR + k];
        a.y  = SF[(rowbase + ml) * LDSTR + k + 1];
        bb.x = Hl[(colbase + ml) * LDSTR + k];
        bb.y = Hl[(colbase + ml) * LDSTR + k + 1];
        acc = wmma4(a, bb, acc);
      }
      for (int v = 0; v < 8; ++v) {
        int row = rowbase + v + 8 * half;
        int col = colbase + ml;
        SIGSH[row * LDSTR + col] = acc[v];
      }
    }
    if (tid < NS) {
      float s = 0.0f;
      for (int k = 0; k < NS; ++k) s += SF[k] * Hl[tid * LDSTR + k];
      SIGSH[tid] = s;
    }
    __syncthreads();

    // ---- 5) xp = Wm @ SF ; zp = Wm @ SH ----
    if (tid < NS) {
      float s = 0.0f;
      for (int k = 0; k < 65; ++k) s += wmw(k) * SF[k * LDSTR + tid];
      xp[tid] = s;
    } else if (tid < 64) {
      int j = tid - 32;
      float s = 0.0f;
      for (int k = 0; k < 65; ++k) s += wmw(k) * SIGSH[k * LDSTR + j];
      zp[j] = s;
    }
    __syncthreads();

    // ---- 6) fused covariance GEMMs (K=65 padded to 68, one tile per wave):
    //         Pp = (Wc*d)^T d + Q ; Pz = (Wc*dz)^T dz + R ; Pxz = (Wc*d)^T dz
    {
      const int r = wave >> 1, c = wave & 1;
      const int mg = 16 * r + ml;
      const int ng = 16 * c + ml;
      v8f accP = {}, accZ = {}, accX = {};
      for (int k0 = 0; k0 < 68; k0 += 4) {
        const int k = k0 + koff;
        v2f aD, bD, aZ, bZ;
        {
          bool in = (k < 65);
          float wc = wcw(k);
          aD.x = in ? wc * (SF[k * LDSTR + mg] - xp[mg]) : 0.0f;
          bD.x = in ? (SF[k * LDSTR + ng] - xp[ng]) : 0.0f;
          aZ.x = in ? wc * (SIGSH[k * LDSTR + mg] - zp[mg]) : 0.0f;
          bZ.x = in ? (SIGSH[k * LDSTR + ng] - zp[ng]) : 0.0f;
        }
        {
          int k1 = k + 1;
          bool in = (k1 < 65);
          float wc = wcw(k1);
          aD.y = in ? wc * (SF[k1 * LDSTR + mg] - xp[mg]) : 0.0f;
          bD.y = in ? (SF[k1 * LDSTR + ng] - xp[ng]) : 0.0f;
          aZ.y = in ? wc * (SIGSH[k1 * LDSTR + mg] - zp[mg]) : 0.0f;
          bZ.y = in ? (SIGSH[k1 * LDSTR + ng] - zp[ng]) : 0.0f;
        }
        accP = wmma4(aD, bD, accP);
        accZ = wmma4(aZ, bZ, accZ);
        accX = wmma4(aD, bZ, accX);
      }
      for (int v = 0; v < 8; ++v) {
        int row = 16 * r + v + 8 * half;
        int col = 16 * c + ml;
        Pp[row * LDSTR + col]  = accP[v] + Qg[row * NS + col];
        Pz[row * LDSTR + col]  = accZ[v] + Rg[row * NS + col];
        Pxz[row * LDSTR + col] = accX[v];
      }
    }
    if (tid >= 64 && tid < 96) {  // innovation residual
      int j = tid - 64;
      resid[j] = Yg[((size_t)b * TT + t) * NS + j] - zp[j];
    }
    __syncthreads();

    // ---- 7) chol(Pz) in place (Pz only needed as its factor from here) ----
    chol32(Pz, invd, tid);

    // ---- 8) solve Pz X = Pxz^T, K = X^T: thread j owns K row j ----
    if (tid < NS) {
#pragma unroll 1
      for (int r2 = 0; r2 < NS; ++r2) {           // forward: Lz w = Pxz[j,:]
        float s = Pxz[tid * LDSTR + r2];
        for (int c2 = 0; c2 < r2; ++c2)
          s -= Pz[r2 * LDSTR + c2] * Km[tid * LDSTR + c2];
        Km[tid * LDSTR + r2] = s * invd[r2];
      }
#pragma unroll 1
      for (int r2 = NS - 1; r2 >= 0; --r2) {      // backward: Lz^T v = w
        float s = Km[tid * LDSTR + r2];
        for (int c2 = r2 + 1; c2 < NS; ++c2)
          s -= Pz[c2 * LDSTR + r2] * Km[tid * LDSTR + c2];
        Km[tid * LDSTR + r2] = s * invd[r2];
      }
    }
    __syncthreads();

    // ---- 9) Pn = Pp - Pxz @ K^T  (since K Pz K^T == Pxz K^T); xn update ----
    {
      const int r = wave >> 1, c = wave & 1;
      v8f acc = {};
      for (int k0 = 0; k0 < NS; k0 += 4) {
        const int k = k0 + koff;
        v2f a, bb;
        a.x  = Pxz[(16 * r + ml) * LDSTR + k];
        a.y  = Pxz[(16 * r + ml) * LDSTR + k + 1];
        bb.x = Km[(16 * c + ml) * LDSTR + k];   // B[k][n] = K[n][k]
        bb.y = Km[(16 * c + ml) * LDSTR + k + 1];
        acc = wmma4(a, bb, acc);
      }
      for (int v = 0; v < 8; ++v) {
        int row = 16 * r + v + 8 * half;
        int col = 16 * c + ml;
        Pm[row * LDSTR + col] = Pp[row * LDSTR + col] - acc[v];
      }
    }
    if (tid < NS) {
      float s = 0.0f;
      for (int k = 0; k < NS; ++k) s += Km[tid * LDSTR + k] * resid[k];
      float xn = xp[tid] + s;
      xv[tid] = xn;
      traj[((size_t)b * 257 + (t + 1)) * NS + tid] = xn;
    }
    __syncthreads();

    // ---- 10) stream Pn to HBM ----
    for (int i = tid; i < NS * NS; i += 128) {
      pk[((size_t)b * 257 + (t + 1)) * (NS * NS) + i] =
          Pm[(i >> 5) * LDSTR + (i & 31)];
    }
    __syncthreads();
  }
}

extern "C" void kernel_launch(void* const* d_in, const int* in_sizes, int n_in,
                              void* d_out, int out_size, void* d_ws,
                              size_t ws_size, hipStream_t stream) {
  (void)n_in; (void)out_size; (void)d_ws; (void)ws_size;
  const float* Y = (const float*)d_in[0];
  const float* F = (const float*)d_in[1];
  const float* H = (const float*)d_in[2];
  const float* Q = (const float*)d_in[3];
  const float* R = (const float*)d_in[4];
  float* out = (float*)d_out;

  const int nTraj = in_sizes[0] / (TT * NS);         // 1024
  float* traj = out;                                  // [N,257,32]
  float* pk   = out + (size_t)nTraj * 257 * NS;       // [N,257,32,32]

  ukf_kernel<<<dim3(nTraj), dim3(128), 0, stream>>>(Y, F, H, Q, R, traj, pk,
                                                    nTraj);
}